// SagPoolBindingNet_72808285602324
// MI455X (gfx1250) — compile-verified
//
#include <hip/hip_runtime.h>
#include <hip/hip_bf16.h>

typedef float v2f __attribute__((ext_vector_type(2)));
typedef float v8f __attribute__((ext_vector_type(8)));
typedef int   v4i __attribute__((ext_vector_type(4)));

#define LDX 132   // leading dim for [64,128] feature buffers (pad vs 64-bank LDS)
#define LDA 68    // leading dim for [64,64] adjacency / norm buffers

#define GAS __attribute__((address_space(1)))
#define LAS __attribute__((address_space(3)))

#if defined(__gfx1250__) && __has_builtin(__builtin_amdgcn_global_load_async_to_lds_b128)
#define USE_ASYNC_LDS 1
#else
#define USE_ASYNC_LDS 0
#endif

// ---- WMMA helpers (16x16 f32 tile, K stepped by 4) -------------------------
// A fragment (16x4, MxK): lanes 0-15 -> rows, VGPR{0,1} = K {0,1}; lanes 16-31 = K {2,3}
// B fragment (4x16, KxN): lanes 0-15 -> cols, VGPR{0,1} = K rows {0,1}; lanes 16-31 = {2,3}
// C (16x16): VGPR v: lanes 0-15 row m0+v, col lane; lanes 16-31 row m0+v+8, col lane-16
__device__ __forceinline__ v8f wmma_tile(const float* __restrict__ A, int lda, int m0,
                                         const float* __restrict__ B, int ldb, int n0,
                                         int K, int lane, v8f acc) {
    const int mr   = m0 + (lane & 15);
    const int nc   = n0 + (lane & 15);
    const int ksel = (lane >> 4) << 1;   // 0 for lanes 0-15, 2 for lanes 16-31
    for (int k0 = 0; k0 < K; k0 += 4) {
        const int kk = k0 + ksel;
        v2f a, b;
        a.x = A[mr * lda + kk];
        a.y = A[mr * lda + kk + 1];
        b.x = B[kk * ldb + nc];
        b.y = B[(kk + 1) * ldb + nc];
        acc = __builtin_amdgcn_wmma_f32_16x16x4_f32(false, a, false, b, (short)0, acc,
                                                    false, false);
    }
    return acc;
}

__device__ __forceinline__ void store_tile_raw(float* __restrict__ C, int ldc, int m0,
                                               int n0, int lane, v8f c) {
    const int col   = n0 + (lane & 15);
    const int rbase = m0 + ((lane >> 4) << 3);
#pragma unroll
    for (int v = 0; v < 8; ++v)
        C[(rbase + v) * ldc + col] = c[v];
}

__device__ __forceinline__ void store_tile_bias_relu(float* __restrict__ C, int ldc, int m0,
                                                     int n0, int lane, v8f c,
                                                     const float* __restrict__ bias) {
    const int col   = n0 + (lane & 15);
    const int rbase = m0 + ((lane >> 4) << 3);
    const float bv  = bias[col];
#pragma unroll
    for (int v = 0; v < 8; ++v)
        C[(rbase + v) * ldc + col] = fmaxf(c[v] + bv, 0.0f);
}

// ---- GCN layer: xbuf <- relu( norm(adjbuf) @ (xbuf @ W) + bias ), n nodes --
__device__ __forceinline__ void gcn_layer(float* xbuf, float* wbuf, float* adjbuf,
                                          float* normbuf, float* dinv,
                                          const float* __restrict__ W, int Kx,
                                          const float* __restrict__ bias,
                                          int n, int tid, int lane, int wave) {
    __syncthreads();
    const int ntiles = (n >> 4) * 8;                 // (n/16) x (128/16)
    // wbuf = xbuf @ W   [n,128]
    for (int t = wave; t < ntiles; t += 8) {
        const int m0 = (t >> 3) << 4, n0 = (t & 7) << 4;
        v8f acc = {};
        acc = wmma_tile(xbuf, LDX, m0, W, 128, n0, Kx, lane, acc);
        store_tile_raw(wbuf, LDX, m0, n0, lane, acc);
    }
    // symmetric normalization of (adj + I)
    if (tid < n) {
        float d = 1.0f;                              // self loop
        for (int j = 0; j < n; ++j) d += adjbuf[tid * LDA + j];
        dinv[tid] = rsqrtf(fmaxf(d, 1e-12f));
    }
    __syncthreads();
    for (int idx = tid; idx < n * n; idx += 256) {
        const int i = idx / n, j = idx - i * n;
        const float a = adjbuf[i * LDA + j] + ((i == j) ? 1.0f : 0.0f);
        normbuf[i * LDA + j] = dinv[i] * a * dinv[j];
    }
    __syncthreads();
    // xbuf = relu(norm @ wbuf + bias)
    for (int t = wave; t < ntiles; t += 8) {
        const int m0 = (t >> 3) << 4, n0 = (t & 7) << 4;
        v8f acc = {};
        acc = wmma_tile(normbuf, LDA, m0, wbuf, LDX, n0, n, lane, acc);
        store_tile_bias_relu(xbuf, LDX, m0, n0, lane, acc, bias);
    }
    __syncthreads();
}

// ---- SAGPool: top-k by GraphConv score, gather x (scaled by tanh) & adj ----
__device__ __forceinline__ void sag_pool(float* xbuf, float* wbuf, float* adjbuf,
                                         float* normbuf, float* tvec, float* uvec,
                                         float* svec, int* perm,
                                         const float* __restrict__ wl,
                                         const float* __restrict__ bl,
                                         const float* __restrict__ wr,
                                         int n, int k, int tid) {
    if (tid < n) {
        float t = 0.0f, u = 0.0f;
        for (int f = 0; f < 128; ++f) {
            const float xv = xbuf[tid * LDX + f];
            t += xv * wl[f];
            u += xv * wr[f];
        }
        tvec[tid] = t;
        uvec[tid] = u;
    }
    __syncthreads();
    if (tid < n) {
        float s = uvec[tid] + bl[0];
        for (int j = 0; j < n; ++j) s += adjbuf[tid * LDA + j] * tvec[j];
        svec[tid] = s;
    }
    __syncthreads();
    // stable descending rank == jax.lax.top_k order
    if (tid < n) {
        const float si = svec[tid];
        int rank = 0;
        for (int j = 0; j < n; ++j) {
            const float sj = svec[j];
            rank += (sj > si) || ((sj == si) && (j < tid));
        }
        if (rank < k) perm[rank] = tid;
    }
    __syncthreads();
    for (int idx = tid; idx < k * 128; idx += 256) {
        const int r = idx >> 7, f = idx & 127;
        const int src = perm[r];
        wbuf[r * LDX + f] = xbuf[src * LDX + f] * tanhf(svec[src]);
    }
    for (int idx = tid; idx < k * k; idx += 256) {
        const int r = idx / k, c = idx - r * k;
        normbuf[r * LDA + c] = adjbuf[perm[r] * LDA + perm[c]];
    }
    __syncthreads();
    for (int idx = tid; idx < k * 128; idx += 256)
        xbuf[(idx >> 7) * LDX + (idx & 127)] = wbuf[(idx >> 7) * LDX + (idx & 127)];
    for (int idx = tid; idx < k * k; idx += 256) {
        const int r = idx / k, c = idx - r * k;
        adjbuf[r * LDA + c] = normbuf[r * LDA + c];
    }
    __syncthreads();
}

__device__ __forceinline__ void readout_add(const float* xbuf, float* racc, int n, int tid) {
    if (tid < 128) {
        float mx = -3.402823466e38f, sm = 0.0f;
        for (int i = 0; i < n; ++i) {
            const float v = xbuf[i * LDX + tid];
            mx = fmaxf(mx, v);
            sm += v;
        }
        racc[tid]       += mx;
        racc[128 + tid] += sm / (float)n;
    }
    __syncthreads();
}

extern "C" __global__ void __launch_bounds__(256)
sagpool_binding_net_kernel(const int* __restrict__ aa, const float* __restrict__ pos,
                           const float* __restrict__ cdr, const float* __restrict__ adj,
                           const float* __restrict__ emb,
                           const float* __restrict__ W1, const float* __restrict__ b1,
                           const float* __restrict__ p1wl, const float* __restrict__ p1bl,
                           const float* __restrict__ p1wr,
                           const float* __restrict__ W2, const float* __restrict__ b2,
                           const float* __restrict__ p2wl, const float* __restrict__ p2bl,
                           const float* __restrict__ p2wr,
                           const float* __restrict__ W3, const float* __restrict__ b3,
                           const float* __restrict__ p3wl, const float* __restrict__ p3bl,
                           const float* __restrict__ p3wr,
                           const float* __restrict__ mW1, const float* __restrict__ mb1,
                           const float* __restrict__ mW2, const float* __restrict__ mb2,
                           const float* __restrict__ mW3, const float* __restrict__ mb3,
                           float* __restrict__ out) {
    __shared__ float xbuf[64 * LDX];
    __shared__ float wbuf[64 * LDX];
    __shared__ float adjbuf[64 * LDA];
    __shared__ float normbuf[64 * LDA];
    __shared__ float w1pad[36 * 128];    // W1 zero-padded from K=34 to K=36
    __shared__ float dinv[64], tvec[64], uvec[64], svec[64];
    __shared__ int   perm[64];
    __shared__ float racc[256];
    __shared__ float h1[128];
    __shared__ float h2[64];

    const int g    = blockIdx.x;
    const int tid  = threadIdx.x;
    const int lane = tid & 31;
    const int wave = tid >> 5;

    racc[tid] = 0.0f;

    // adjacency tile (16 KB/graph, the dominant HBM traffic)
#if USE_ASYNC_LDS
    // async global->LDS DMA (no VGPR staging), tracked on ASYNCcnt
    for (int idx = tid; idx < 64 * 16; idx += 256) {
        const int i = idx >> 4, q = idx & 15;
        float* src = const_cast<float*>(adj) + (size_t)g * 4096 + (size_t)i * 64 + (q << 2);
        float* dst = &adjbuf[i * LDA + (q << 2)];
        __builtin_amdgcn_global_load_async_to_lds_b128(
            (GAS v4i*)(GAS void*)src,
            (LAS v4i*)(LAS void*)dst, 0, 0);
    }
#else
    for (int idx = tid; idx < 64 * 16; idx += 256) {
        const int i = idx >> 4, q = idx & 15;
        const float4 v =
            reinterpret_cast<const float4*>(adj + (size_t)g * 4096 + (size_t)i * 64)[q];
        float* dst = &adjbuf[i * LDA + (q << 2)];
        dst[0] = v.x; dst[1] = v.y; dst[2] = v.z; dst[3] = v.w;
    }
#endif

    // stage W1 (padded) into LDS so the WMMA B-loads never run out of bounds
    for (int idx = tid; idx < 36 * 128; idx += 256) {
        const int r = idx >> 7, c = idx & 127;
        w1pad[idx] = (r < 34) ? W1[r * 128 + c] : 0.0f;
    }
    // x0 = [emb[aa], pos, is_cdr3, 0, 0]  (IN=34 padded to 36)
    for (int idx = tid; idx < 64 * 36; idx += 256) {
        const int i = idx / 36, c = idx - i * 36;
        float v;
        if (c < 32)       v = emb[aa[g * 64 + i] * 32 + c];
        else if (c == 32) v = pos[g * 64 + i];
        else if (c == 33) v = cdr[g * 64 + i];
        else              v = 0.0f;
        xbuf[i * LDX + c] = v;
    }
#if USE_ASYNC_LDS
    asm volatile("s_wait_asynccnt 0" ::: "memory");
#endif
    __syncthreads();

    // ---- stage 1 ----
    gcn_layer(xbuf, wbuf, adjbuf, normbuf, dinv, w1pad, 36, b1, 64, tid, lane, wave);
    sag_pool(xbuf, wbuf, adjbuf, normbuf, tvec, uvec, svec, perm, p1wl, p1bl, p1wr,
             64, 32, tid);
    readout_add(xbuf, racc, 32, tid);
    // ---- stage 2 ----
    gcn_layer(xbuf, wbuf, adjbuf, normbuf, dinv, W2, 128, b2, 32, tid, lane, wave);
    sag_pool(xbuf, wbuf, adjbuf, normbuf, tvec, uvec, svec, perm, p2wl, p2bl, p2wr,
             32, 16, tid);
    readout_add(xbuf, racc, 16, tid);
    // ---- stage 3 ----
    gcn_layer(xbuf, wbuf, adjbuf, normbuf, dinv, W3, 128, b3, 16, tid, lane, wave);
    sag_pool(xbuf, wbuf, adjbuf, normbuf, tvec, uvec, svec, perm, p3wl, p3bl, p3wr,
             16, 8, tid);
    readout_add(xbuf, racc, 8, tid);

    // ---- MLP head ----
    if (tid < 128) {
        float h = mb1[tid];
        for (int k = 0; k < 256; ++k) h += racc[k] * mW1[k * 128 + tid];
        h1[tid] = fmaxf(h, 0.0f);
    }
    __syncthreads();
    if (tid < 64) {
        float h = mb2[tid];
        for (int k = 0; k < 128; ++k) h += h1[k] * mW2[k * 64 + tid];
        h2[tid] = fmaxf(h, 0.0f);
    }
    __syncthreads();
    if (tid == 0) {
        float o = mb3[0];
        for (int k = 0; k < 64; ++k) o += h2[k] * mW3[k];
        out[g] = o;
    }
}

extern "C" void kernel_launch(void* const* d_in, const int* in_sizes, int n_in,
                              void* d_out, int out_size, void* d_ws, size_t ws_size,
                              hipStream_t stream) {
    const int*   aa   = (const int*)d_in[0];
    const float* pos  = (const float*)d_in[1];
    const float* cdr  = (const float*)d_in[2];
    const float* adj  = (const float*)d_in[3];
    const float* emb  = (const float*)d_in[4];
    const float* W1   = (const float*)d_in[5];
    const float* b1   = (const float*)d_in[6];
    const float* p1wl = (const float*)d_in[7];
    const float* p1bl = (const float*)d_in[8];
    const float* p1wr = (const float*)d_in[9];
    const float* W2   = (const float*)d_in[10];
    const float* b2   = (const float*)d_in[11];
    const float* p2wl = (const float*)d_in[12];
    const float* p2bl = (const float*)d_in[13];
    const float* p2wr = (const float*)d_in[14];
    const float* W3   = (const float*)d_in[15];
    const float* b3   = (const float*)d_in[16];
    const float* p3wl = (const float*)d_in[17];
    const float* p3bl = (const float*)d_in[18];
    const float* p3wr = (const float*)d_in[19];
    const float* mW1  = (const float*)d_in[20];
    const float* mb1  = (const float*)d_in[21];
    const float* mW2  = (const float*)d_in[22];
    const float* mb2  = (const float*)d_in[23];
    const float* mW3  = (const float*)d_in[24];
    const float* mb3  = (const float*)d_in[25];
    float* out = (float*)d_out;

    const int B = in_sizes[0] / 64;   // 4096 graphs, one workgroup each
    hipLaunchKernelGGL(sagpool_binding_net_kernel, dim3(B), dim3(256), 0, stream,
                       aa, pos, cdr, adj, emb, W1, b1, p1wl, p1bl, p1wr,
                       W2, b2, p2wl, p2bl, p2wr, W3, b3, p3wl, p3bl, p3wr,
                       mW1, mb1, mW2, mb2, mW3, mb3, out);
}